// SSL_11716670784252
// MI455X (gfx1250) — compile-verified
//
#include <hip/hip_runtime.h>
#include <hip/hip_bf16.h>

#define N_USERS 50000
#define N_ITEMS 50000
#define N_NODES 100000
#define DIM     64
#define BATCH   2048
#define SSL_TEMP 0.5f
#define SSL_REG  0.5f
#define INV_TEMP (1.0f / SSL_TEMP)

// WMMA tiling: 2048 batch rows = 128 tiles/side -> 256 M-tiles total.
// 50000 rows/side = 3125 N-tiles = 25 chunks x 125 tiles.
#define MTILES_PER_SIDE 128
#define NTILES_PER_SIDE 3125
#define NCHUNKS         25
#define TILES_PER_CHUNK 125

typedef __attribute__((ext_vector_type(2))) float v2f;
typedef __attribute__((ext_vector_type(8))) float v8f;

__device__ __forceinline__ const float* ego_row(const float* ue, const float* ie, int n) {
    return (n < N_USERS) ? (ue + (size_t)n * DIM) : (ie + (size_t)(n - N_USERS) * DIM);
}

// ---- mark which rows of e1 are actually needed (batch rows only) ----
__global__ void k_flags(const int* __restrict__ nu, const int* __restrict__ ni,
                        int* __restrict__ flags) {
    int b = blockIdx.x * 256 + threadIdx.x;
    if (b < BATCH)            flags[nu[b]] = 1;
    else if (b < 2 * BATCH)   flags[N_USERS + ni[b - BATCH]] = 1;
}

// ---- COO SpMM: acc[row] += val * ego[col]; optional row-flag gating ----
__global__ void k_spmm(const int* __restrict__ rows, const int* __restrict__ cols,
                       const float* __restrict__ vals,
                       const float* __restrict__ ue, const float* __restrict__ ie,
                       const int* __restrict__ flags, float* __restrict__ acc,
                       int nedges) {
    int gid = blockIdx.x * 256 + threadIdx.x;
    int e = gid >> 4;
    if (e >= nedges) return;
    int r = rows[e];
    if (flags && !flags[r]) return;          // skip gathers for unused e1 rows
    int d = (gid & 15) * 4;
    int c = cols[e];
    float v = vals[e];
    const float4 x = *(const float4*)(ego_row(ue, ie, c) + d);
    float* a = acc + (size_t)r * DIM + d;
    atomicAdd(a + 0, v * x.x);
    atomicAdd(a + 1, v * x.y);
    atomicAdd(a + 2, v * x.z);
    atomicAdd(a + 3, v * x.w);
}

// ---- zn2 = l2norm(0.5*(ego + acc2)) for all 100000 rows; 1 wave per row ----
__global__ void k_norm_all(const float* __restrict__ ue, const float* __restrict__ ie,
                           const float* __restrict__ acc2, float* __restrict__ zn2) {
    int row  = blockIdx.x * 8 + (threadIdx.x >> 5);
    int lane = threadIdx.x & 31;
    if (row >= N_NODES) return;
    const float* x = ego_row(ue, ie, row);
    const float* a = acc2 + (size_t)row * DIM;
    float ex = 0.5f * (x[2 * lane]     + a[2 * lane]);
    float ey = 0.5f * (x[2 * lane + 1] + a[2 * lane + 1]);
    float ss = ex * ex + ey * ey;
    for (int m = 16; m; m >>= 1) ss += __shfl_xor(ss, m, 32);
    float inv = 1.0f / fmaxf(sqrtf(ss), 1e-12f);
    zn2[(size_t)row * DIM + 2 * lane]     = ex * inv;
    zn2[(size_t)row * DIM + 2 * lane + 1] = ey * inv;
}

// ---- z1b = l2norm(e1[batch rows]); dotb = z1 . zn2[row]  (positive pair) ----
__global__ void k_batch_z1(const float* __restrict__ ue, const float* __restrict__ ie,
                           const float* __restrict__ acc1,
                           const int* __restrict__ nu, const int* __restrict__ ni,
                           const float* __restrict__ zn2,
                           float* __restrict__ z1b, float* __restrict__ dotb) {
    int b    = blockIdx.x * 8 + (threadIdx.x >> 5);
    int lane = threadIdx.x & 31;
    if (b >= 2 * BATCH) return;
    int row = (b < BATCH) ? nu[b] : (N_USERS + ni[b - BATCH]);
    const float* x = ego_row(ue, ie, row);
    const float* a = acc1 + (size_t)row * DIM;
    float ex = 0.5f * (x[2 * lane]     + a[2 * lane]);
    float ey = 0.5f * (x[2 * lane + 1] + a[2 * lane + 1]);
    float ss = ex * ex + ey * ey;
    for (int m = 16; m; m >>= 1) ss += __shfl_xor(ss, m, 32);
    float inv = 1.0f / fmaxf(sqrtf(ss), 1e-12f);
    float zx = ex * inv, zy = ey * inv;
    z1b[(size_t)b * DIM + 2 * lane]     = zx;
    z1b[(size_t)b * DIM + 2 * lane + 1] = zy;
    float d = zx * zn2[(size_t)row * DIM + 2 * lane]
            + zy * zn2[(size_t)row * DIM + 2 * lane + 1];
    for (int m = 16; m; m >>= 1) d += __shfl_xor(d, m, 32);
    if (lane == 0) dotb[b] = d;
}

// ---- WMMA GEMM + exp row-sums: ttl[m] += sum_n exp((z1[m].zn2[n]) / temp) ----
// grid.x = 256 M-tiles (128 user + 128 item), grid.y = 25 N-chunks, block = 1 wave.
__global__ void k_nce_gemm(const float* __restrict__ z1b, const float* __restrict__ zn2,
                           float* __restrict__ ttl) {
    const int lane = threadIdx.x;        // 0..31, EXEC all-ones (no divergence)
    const int half = lane >> 4;          // which K-pair / which M half for C/D
    const int l16  = lane & 15;
    const int mt   = blockIdx.x;

    // A fragment: 16x4 f32 layout -> lane holds M=l16, K = {2*half + 4*kk, +1}
    const float* arow = z1b + (size_t)(mt * 16 + l16) * DIM + 2 * half;
    v2f afrag[16];
#pragma unroll
    for (int kk = 0; kk < 16; ++kk) {
        afrag[kk].x = arow[4 * kk];
        afrag[kk].y = arow[4 * kk + 1];
    }

    const int jbase = (mt < MTILES_PER_SIDE) ? 0 : N_USERS;   // user vs item side
    const int t0 = blockIdx.y * TILES_PER_CHUNK;

    float rowsum[8] = {0.f, 0.f, 0.f, 0.f, 0.f, 0.f, 0.f, 0.f};

    for (int t = t0; t < t0 + TILES_PER_CHUNK; ++t) {
        // B fragment (B = zall^T): lane holds N=l16, K = {2*half + 4*kk, +1}
        const float* brow = zn2 + (size_t)(jbase + t * 16 + l16) * DIM + 2 * half;
        if (t + 1 < t0 + TILES_PER_CHUNK)
            __builtin_prefetch(zn2 + (size_t)(jbase + (t + 1) * 16 + l16) * DIM, 0, 3);
        v8f acc = {};
#pragma unroll
        for (int kk = 0; kk < 16; ++kk) {
            v2f bfrag;
            bfrag.x = brow[4 * kk];
            bfrag.y = brow[4 * kk + 1];
            acc = __builtin_amdgcn_wmma_f32_16x16x4_f32(
                false, afrag[kk], false, bfrag, (short)0, acc, false, false);
        }
        // C/D layout: lane(l16,half) VGPR v -> D[m = v + 8*half][n = l16]
#pragma unroll
        for (int v = 0; v < 8; ++v) rowsum[v] += __expf(acc[v] * INV_TEMP);
    }

    // reduce over the 16 lanes of each half (n-dimension), then atomically
    // accumulate into the per-batch-row total.
#pragma unroll
    for (int v = 0; v < 8; ++v) {
        float s = rowsum[v];
        s += __shfl_xor(s, 1, 32);
        s += __shfl_xor(s, 2, 32);
        s += __shfl_xor(s, 4, 32);
        s += __shfl_xor(s, 8, 32);
        if (l16 == 0) atomicAdd(&ttl[mt * 16 + v + 8 * half], s);
    }
}

// ---- final: out = SSL_REG * sum_b (log ttl[b] - dot[b]/temp) ----
__global__ void k_finish(const float* __restrict__ ttl, const float* __restrict__ dotb,
                         float* __restrict__ out) {
    __shared__ float sbuf[32];
    int tid = threadIdx.x;                // 1024 threads = 32 waves
    float s = 0.f;
    for (int b = tid; b < 2 * BATCH; b += 1024)
        s += __logf(ttl[b]) - dotb[b] * INV_TEMP;
    for (int m = 16; m; m >>= 1) s += __shfl_xor(s, m, 32);
    if ((tid & 31) == 0) sbuf[tid >> 5] = s;
    __syncthreads();
    if (tid < 32) {
        float t = sbuf[tid];
        for (int m = 16; m; m >>= 1) t += __shfl_xor(t, m, 32);
        if (tid == 0) out[0] = SSL_REG * t;
    }
}

extern "C" void kernel_launch(void* const* d_in, const int* in_sizes, int n_in,
                              void* d_out, int out_size, void* d_ws, size_t ws_size,
                              hipStream_t stream) {
    const float* user_emb = (const float*)d_in[0];
    const float* item_emb = (const float*)d_in[1];
    const int*   rows1    = (const int*)d_in[2];
    const int*   cols1    = (const int*)d_in[3];
    const float* vals1    = (const float*)d_in[4];
    const int*   rows2    = (const int*)d_in[5];
    const int*   cols2    = (const int*)d_in[6];
    const float* vals2    = (const float*)d_in[7];
    const int*   nodes_u  = (const int*)d_in[8];
    const int*   nodes_i  = (const int*)d_in[9];
    const int nedges1 = in_sizes[2];
    const int nedges2 = in_sizes[5];

    // workspace layout
    float* acc1 = (float*)d_ws;                       // N_NODES*DIM
    float* acc2 = acc1 + (size_t)N_NODES * DIM;       // N_NODES*DIM
    float* zn2  = acc2 + (size_t)N_NODES * DIM;       // N_NODES*DIM
    float* z1b  = zn2  + (size_t)N_NODES * DIM;       // 2*BATCH*DIM
    float* dotb = z1b  + (size_t)2 * BATCH * DIM;     // 2*BATCH
    float* ttl  = dotb + 2 * BATCH;                   // 2*BATCH
    int*   flags = (int*)(ttl + 2 * BATCH);           // N_NODES

    hipMemsetAsync(acc1, 0, (size_t)2 * N_NODES * DIM * sizeof(float), stream); // acc1+acc2
    hipMemsetAsync(ttl,  0, (size_t)2 * BATCH * sizeof(float), stream);
    hipMemsetAsync(flags, 0, (size_t)N_NODES * sizeof(int), stream);

    k_flags<<<(2 * BATCH + 255) / 256, 256, 0, stream>>>(nodes_u, nodes_i, flags);

    int thr2 = nedges2 * 16;
    k_spmm<<<(thr2 + 255) / 256, 256, 0, stream>>>(rows2, cols2, vals2,
                                                   user_emb, item_emb,
                                                   nullptr, acc2, nedges2);
    int thr1 = nedges1 * 16;
    k_spmm<<<(thr1 + 255) / 256, 256, 0, stream>>>(rows1, cols1, vals1,
                                                   user_emb, item_emb,
                                                   flags, acc1, nedges1);

    k_norm_all<<<(N_NODES + 7) / 8, 256, 0, stream>>>(user_emb, item_emb, acc2, zn2);

    k_batch_z1<<<(2 * BATCH + 7) / 8, 256, 0, stream>>>(user_emb, item_emb, acc1,
                                                        nodes_u, nodes_i, zn2,
                                                        z1b, dotb);

    dim3 ggrid(2 * MTILES_PER_SIDE, NCHUNKS);
    k_nce_gemm<<<ggrid, 32, 0, stream>>>(z1b, zn2, ttl);

    k_finish<<<1, 1024, 0, stream>>>(ttl, dotb, (float*)d_out);
}